// SoftDTWLoss_44263932953213
// MI455X (gfx1250) — compile-verified
//
#include <hip/hip_runtime.h>
#include <hip/hip_bf16.h>

// Soft-DTW (gamma = 1.0) for x:(64,8,512) f32, y:(64,8,512) f32 -> scalar mean cost.
//
// Path A (workspace large enough):
//   1) norms_kernel : xn[b,t] = sum_c x^2, yn[b,u] = sum_c y^2
//   2) gemm_D       : D[b,t,u] = xn + yn - 2*x.y  via V_WMMA_F32_16X16X4_F32
//   3) dtw_dp       : anti-diagonal wavefront DP (D load software-pipelined),
//                     one workgroup per batch
//   4) mean_kernel  : deterministic mean over 64 batch costs
// Path B (tiny workspace): fused DP recomputing d on the fly from LDS-resident x and register-resident y.

#define B_DIM 64
#define C_DIM 8
#define T_DIM 512
#define U_DIM 512

__device__ float g_cost[B_DIM];  // per-batch costs; fully overwritten every launch

typedef float v2f __attribute__((ext_vector_type(2)));
typedef float v8f __attribute__((ext_vector_type(8)));

// ---------------------------------------------------------------- norms
__global__ void norms_kernel(const float* __restrict__ x, const float* __restrict__ y,
                             float* __restrict__ xn, float* __restrict__ yn) {
    int i = blockIdx.x * blockDim.x + threadIdx.x;       // 0 .. B*T-1
    if (i >= B_DIM * T_DIM) return;
    int b = i >> 9;
    int t = i & (T_DIM - 1);
    float sx = 0.0f, sy = 0.0f;
#pragma unroll
    for (int c = 0; c < C_DIM; ++c) {
        float xv = x[((size_t)b * C_DIM + c) * T_DIM + t];
        float yv = y[((size_t)b * C_DIM + c) * U_DIM + t];
        sx += xv * xv;
        sy += yv * yv;
    }
    xn[i] = sx;
    yn[i] = sy;
}

// ---------------------------------------------------------------- D via WMMA f32
// One wave per 16x16 tile of D[b]. K = C_DIM = 8 -> two chained 16x16x4 f32 WMMAs.
__global__ void gemm_D(const float* __restrict__ x, const float* __restrict__ y,
                       const float* __restrict__ xn, const float* __restrict__ yn,
                       float* __restrict__ D) {
    const int lane = threadIdx.x & 31;
    const int wave = (blockIdx.x * blockDim.x + threadIdx.x) >> 5;  // global wave id
    const int tilesPerBatch = (T_DIM / 16) * (U_DIM / 16);          // 1024
    const int b = wave / tilesPerBatch;
    const int tile = wave % tilesPerBatch;
    const int ti = tile / (U_DIM / 16);
    const int tj = tile % (U_DIM / 16);
    const int t0 = ti * 16, u0 = tj * 16;
    const int half = lane >> 4;      // 0: lanes 0-15, 1: lanes 16-31
    const int l = lane & 15;

    const float* xb = x + (size_t)b * C_DIM * T_DIM;
    const float* yb = y + (size_t)b * C_DIM * U_DIM;

    v8f acc = {};
#pragma unroll
    for (int k0 = 0; k0 < C_DIM; k0 += 4) {
        // f32 A 16x4 layout: lane half selects K pair {k0,k0+1} vs {k0+2,k0+3}, M = l
        v2f av, bv;
        av.x = xb[(size_t)(k0 + 2 * half + 0) * T_DIM + t0 + l];
        av.y = xb[(size_t)(k0 + 2 * half + 1) * T_DIM + t0 + l];
        // f32 B 4x16 layout mirrors A with lane <-> N
        bv.x = yb[(size_t)(k0 + 2 * half + 0) * U_DIM + u0 + l];
        bv.y = yb[(size_t)(k0 + 2 * half + 1) * U_DIM + u0 + l];
        acc = __builtin_amdgcn_wmma_f32_16x16x4_f32(
            /*neg_a=*/false, av, /*neg_b=*/false, bv,
            /*c_mod=*/(short)0, acc, /*reuse_a=*/false, /*reuse_b=*/false);
    }

    // C/D layout: VGPR r, lane -> M = r + 8*half, N = l
    const float ynv = yn[(size_t)b * U_DIM + u0 + l];
    float* Db = D + (size_t)b * T_DIM * U_DIM;
#pragma unroll
    for (int r = 0; r < 8; ++r) {
        const int m = r + 8 * half;
        const float xv = xn[(size_t)b * T_DIM + t0 + m];
        Db[(size_t)(t0 + m) * U_DIM + (u0 + l)] = xv + ynv - 2.0f * acc[r];
    }
}

// ---------------------------------------------------------------- softmin3 (gamma=1)
__device__ __forceinline__ float softmin3(float a, float b, float c) {
    float m = fminf(a, fminf(b, c));
    // m is finite for every reachable cell; exp(m - inf) = 0 is fine.
    float s = __expf(m - a) + __expf(m - b) + __expf(m - c);
    return m - __logf(s);   // s in [1,3] -> log well conditioned
}

// ---------------------------------------------------------------- DP from precomputed D
// One workgroup (512 threads = 16 wave32) per batch. The global D load for
// diagonal s+1 is prefetched during diagonal s so its L2 latency overlaps the
// softmin + barrier on the serial critical path.
__global__ void __launch_bounds__(U_DIM) dtw_dp(const float* __restrict__ D) {
    const int b = blockIdx.x;
    const int uu = threadIdx.x;           // column u = uu + 1
    const int u = uu + 1;
    const float INF = __builtin_inff();

    __shared__ float buf[3][U_DIM + 1];
    for (int i = threadIdx.x; i < 3 * (U_DIM + 1); i += blockDim.x)
        (&buf[0][0])[i] = INF;
    __syncthreads();
    if (threadIdx.x == 0) buf[0][0] = 0.0f;   // R[0][0] on diagonal 0
    __syncthreads();

    // This thread's D column base: element (t-1, u-1) is dcol[(t-1) * U_DIM].
    const float* dcol = D + (size_t)b * T_DIM * U_DIM + (u - 1);
    float d_next = dcol[0];                   // prefetch for first active diag (t = 1)
    int p2 = 0, p1 = 1, cu = 2;               // diagonals s-2, s-1, s

    for (int s = 2; s <= T_DIM + U_DIM; ++s) {
        const int t = s - u;
        if (t >= 1 && t <= T_DIM) {
            const float d_cur = d_next;
            if (t < T_DIM) d_next = dcol[(size_t)t * U_DIM];   // prefetch diag s+1
            const float up   = buf[p1][u];       // R[t-1][u]
            const float left = buf[p1][u - 1];   // R[t][u-1]
            const float dg   = buf[p2][u - 1];   // R[t-1][u-1]
            const float r = d_cur + softmin3(up, dg, left);
            buf[cu][u] = r;
            if (s == T_DIM + U_DIM) g_cost[b] = r;   // final cell (T,U)
        } else if (t == 0) {
            buf[cu][u] = INF;                    // R[0][u] boundary
        }
        if (threadIdx.x == 0) buf[cu][0] = INF;  // R[s][0] boundary
        __syncthreads();
        int tmp = p2; p2 = p1; p1 = cu; cu = tmp;
    }
}

// ---------------------------------------------------------------- fused DP fallback
__global__ void __launch_bounds__(U_DIM) dtw_fused(const float* __restrict__ x,
                                                   const float* __restrict__ y) {
    const int b = blockIdx.x;
    const int uu = threadIdx.x;
    const int u = uu + 1;
    const float INF = __builtin_inff();

    __shared__ float buf[3][U_DIM + 1];
    __shared__ float xs[C_DIM][T_DIM];   // 16 KB: whole x[b]
    __shared__ float xn[T_DIM];

    float yreg[C_DIM];
    float ynv = 0.0f, xnv = 0.0f;
#pragma unroll
    for (int c = 0; c < C_DIM; ++c) {
        float xv = x[((size_t)b * C_DIM + c) * T_DIM + uu];
        xs[c][uu] = xv;
        xnv += xv * xv;
        float yv = y[((size_t)b * C_DIM + c) * U_DIM + uu];
        yreg[c] = yv;
        ynv += yv * yv;
    }
    xn[uu] = xnv;
    for (int i = threadIdx.x; i < 3 * (U_DIM + 1); i += blockDim.x)
        (&buf[0][0])[i] = INF;
    __syncthreads();
    if (threadIdx.x == 0) buf[0][0] = 0.0f;
    __syncthreads();

    int p2 = 0, p1 = 1, cu = 2;
    for (int s = 2; s <= T_DIM + U_DIM; ++s) {
        const int t = s - u;
        if (t >= 1 && t <= T_DIM) {
            const float up   = buf[p1][u];
            const float left = buf[p1][u - 1];
            const float dg   = buf[p2][u - 1];
            float dot = 0.0f;
#pragma unroll
            for (int c = 0; c < C_DIM; ++c) dot += xs[c][t - 1] * yreg[c];
            const float d = xn[t - 1] + ynv - 2.0f * dot;
            const float r = d + softmin3(up, dg, left);
            buf[cu][u] = r;
            if (s == T_DIM + U_DIM) g_cost[b] = r;
        } else if (t == 0) {
            buf[cu][u] = INF;
        }
        if (threadIdx.x == 0) buf[cu][0] = INF;
        __syncthreads();
        int tmp = p2; p2 = p1; p1 = cu; cu = tmp;
    }
}

// ---------------------------------------------------------------- mean (deterministic)
__global__ void mean_kernel(float* __restrict__ out) {
    if (threadIdx.x == 0 && blockIdx.x == 0) {
        float s = 0.0f;
        for (int i = 0; i < B_DIM; ++i) s += g_cost[i];
        *out = s / (float)B_DIM;
    }
}

// ---------------------------------------------------------------- host
extern "C" void kernel_launch(void* const* d_in, const int* in_sizes, int n_in,
                              void* d_out, int out_size, void* d_ws, size_t ws_size,
                              hipStream_t stream) {
    const float* x = (const float*)d_in[0];
    const float* y = (const float*)d_in[1];
    float* out = (float*)d_out;

    const size_t D_elems = (size_t)B_DIM * T_DIM * U_DIM;            // 16,777,216
    const size_t need = (D_elems + 2 * (size_t)B_DIM * T_DIM) * sizeof(float);

    if (ws_size >= need) {
        float* D  = (float*)d_ws;
        float* xn = D + D_elems;
        float* yn = xn + (size_t)B_DIM * T_DIM;

        norms_kernel<<<(B_DIM * T_DIM + 255) / 256, 256, 0, stream>>>(x, y, xn, yn);

        // 64 batches * 1024 tiles = 65536 waves; 8 waves (256 threads) per block
        const int totalWaves = B_DIM * (T_DIM / 16) * (U_DIM / 16);
        gemm_D<<<totalWaves / 8, 256, 0, stream>>>(x, y, xn, yn, D);

        dtw_dp<<<B_DIM, U_DIM, 0, stream>>>(D);
    } else {
        dtw_fused<<<B_DIM, U_DIM, 0, stream>>>(x, y);
    }
    mean_kernel<<<1, 1, 0, stream>>>(out);
}